// DiagPooling_78477642433377
// MI455X (gfx1250) — compile-verified
//
#include <hip/hip_runtime.h>
#include <stdint.h>

#define AS1 __attribute__((address_space(1)))
#define AS3 __attribute__((address_space(3)))

typedef __attribute__((ext_vector_type(4))) int v4i;

namespace {
constexpr int kB       = 8;
constexpr int kC       = 128;
constexpr int kH       = 512;
constexpr int kW       = 512;
constexpr int kSeg     = kH + kW - 1;   // 1023 diagonals
constexpr int kOff     = kH + 1;        // 513 requested offsets (-256..256)
constexpr int kPanel   = 8;             // rows per LDS panel (16 KB)
constexpr int kNP      = kH / kPanel;   // 64 panels per plane
constexpr int kThreads = 512;
constexpr int kBufs    = 4;             // async pipeline depth (3 panels ahead)
}

// ---------------- CDNA5 async global->LDS helpers ----------------
__device__ __forceinline__ void async_copy16(const float* g, float* l) {
#if __has_builtin(__builtin_amdgcn_global_load_async_to_lds_b128)
  __builtin_amdgcn_global_load_async_to_lds_b128(
      (AS1 v4i*)(uintptr_t)g,                    // global source (AS1, v4i*)
      (AS3 v4i*)(uint32_t)(uintptr_t)l,          // LDS dest: low 32 bits of generic addr
      0, 0);
#else
  uint32_t loff  = (uint32_t)(uintptr_t)l;
  uint64_t gaddr = (uint64_t)(uintptr_t)g;
  asm volatile("global_load_async_to_lds_b128 %0, %1, off"
               :: "v"(loff), "v"(gaddr) : "memory");
#endif
}

#if __has_builtin(__builtin_amdgcn_s_wait_asynccnt)
  #define ASYNC_WAIT(N) __builtin_amdgcn_s_wait_asynccnt(N)
#else
  #define ASYNC_WAIT(N) asm volatile("s_wait_asynccnt " #N ::: "memory")
#endif

// Issue one 8x512-f32 panel (16KB) as 2 async b128 loads per thread (ASYNCcnt += 2/wave).
__device__ __forceinline__ void issue_panel(const float* __restrict__ plane,
                                            int p, float* __restrict__ buf, int t) {
  const float* g = plane + (size_t)p * kPanel * kW;
#pragma unroll
  for (int k = 0; k < 2; ++k) {
    const int fo = (t + k * 512) * 4;   // float index, 16B-aligned
    async_copy16(g + fo, buf + fo);
  }
}

// One block per (b,c) plane. Thread t accumulates diagonals t+511 (unwrapped)
// and t-1 (wrapped) over all 512 rows via shifted reads from LDS-staged panels.
// 4-deep async double... quad buffering: panels p+1..p+3 stream while p is consumed.
__global__ __launch_bounds__(kThreads)
void diag_sum_kernel(const float* __restrict__ x, float* __restrict__ gacc) {
  __shared__ float smem[kBufs][kPanel * kW];   // 64 KB
  const int t  = threadIdx.x;                  // 0..511 = column shift id
  const int bc = blockIdx.x;                   // b*kC + c
  const float* plane = x + (size_t)bc * kH * kW;
  const int b = bc / kC;

  float acc_hi = 0.0f;   // diagonal id t + 511
  float acc_lo = 0.0f;   // diagonal id t - 1  (only t >= 1)

#pragma unroll
  for (int q = 0; q < kBufs - 1; ++q)          // prefetch panels 0..2
    issue_panel(plane, q, &smem[q][0], t);

  for (int p = 0; p < kNP; ++p) {
    if (p + kBufs - 1 < kNP)                   // keep 3 panels in flight
      issue_panel(plane, p + kBufs - 1, &smem[(p + kBufs - 1) % kBufs][0], t);

    // In-order async completion: wait until only the panels beyond p remain.
    const int ahead = (kNP - 1 - p < kBufs - 1) ? (kNP - 1 - p) : (kBufs - 1);
    if      (ahead == 3) ASYNC_WAIT(6);
    else if (ahead == 2) ASYNC_WAIT(4);
    else if (ahead == 1) ASYNC_WAIT(2);
    else                 ASYNC_WAIT(0);
    __syncthreads();                           // every wave waited -> panel p in LDS

    const float* buf = &smem[p % kBufs][0];
    const int i0 = p * kPanel;
#pragma unroll
    for (int l = 0; l < kPanel; ++l) {
      const int i = i0 + l;
      const int s = i + t;                           // shifted column
      const float v = buf[l * kW + (s & (kW - 1))];  // lanes hit distinct banks
      if (s < kW) acc_hi += v; else acc_lo += v;
    }
    __syncthreads();   // protect buffer (p%kBufs) before it is reissued at p+kBufs
  }

  float* acc_b = gacc + (size_t)b * kSeg;
  unsafeAtomicAdd(&acc_b[t + (kH - 1)], acc_hi);   // native global_atomic_add_f32
  if (t > 0) unsafeAtomicAdd(&acc_b[t - 1], acc_lo);
}

__global__ void zero_kernel(float* __restrict__ p, int n) {
  int i = blockIdx.x * blockDim.x + threadIdx.x;
  if (i < n) p[i] = 0.0f;
}

__global__ void finalize_kernel(const float* __restrict__ gacc, float* __restrict__ out) {
  int k = blockIdx.x * blockDim.x + threadIdx.x;
  if (k >= kB * kOff) return;
  const int b   = k / kOff;
  const int o   = k - b * kOff;
  const int off = o - kH / 2;                    // -256..256
  const int idx = off + (kH - 1);                // diagonal id
  const int al  = off < 0 ? -off : off;
  const float dlen  = (float)(kH - al);
  const float denom = (float)kC * dlen;          // mean over (C, diag_len)
  out[(size_t)b * kOff + o] = gacc[(size_t)b * kSeg + idx] / denom;
}

// ---------------- host launcher ----------------
extern "C" void kernel_launch(void* const* d_in, const int* in_sizes, int n_in,
                              void* d_out, int out_size, void* d_ws, size_t ws_size,
                              hipStream_t stream) {
  (void)in_sizes; (void)n_in; (void)out_size; (void)ws_size;
  const float* x = (const float*)d_in[0];
  float* out  = (float*)d_out;
  float* gacc = (float*)d_ws;                    // kB * kSeg floats (32 KB)

  const int nacc = kB * kSeg;
  zero_kernel<<<(nacc + 255) / 256, 256, 0, stream>>>(gacc, nacc);
  diag_sum_kernel<<<kB * kC, kThreads, 0, stream>>>(x, gacc);
  const int nout = kB * kOff;
  finalize_kernel<<<(nout + 255) / 256, 256, 0, stream>>>(gacc, out);
}